// Sampler_72035191488970
// MI455X (gfx1250) — compile-verified
//
#include <hip/hip_runtime.h>
#include <hip/hip_bf16.h>
#include <stdint.h>

#define BD   512                 // threads per block (16 wave32)
#define CH   2048                // chunk elements staged per double-buffer slot (BD*4)
#define VSZ  128000
#define BSZ  128

// ---- order-preserving float <-> uint32 key (larger float => larger key) ----
__device__ __forceinline__ uint32_t fkey(float f) {
  uint32_t u = __float_as_uint(f);
  return (u & 0x80000000u) ? ~u : (u | 0x80000000u);
}
__device__ __forceinline__ float funkey(uint32_t k) {
  uint32_t u = (k & 0x80000000u) ? (k ^ 0x80000000u) : ~k;
  return __uint_as_float(u);
}

struct SMem {
  alignas(16) float stage[2][CH];      // 16 KB async-DMA staging
  unsigned long long skey[BD];         // 4 KB candidate sort keys
  float redM[BD];                      // logsumexp reduce (max)
  float redS[BD];                      // logsumexp reduce (sum)
  unsigned cntK[256];                  // count histogram (top-64 radix select)
  unsigned cntP[256];                  // count histogram (top-p boundary ties)
  float    wP[256];                    // prob-mass histogram (top-p radix select)
  float Lmax, Dsum, invD;
  unsigned prefK, prefP, rK;
  float Aout, sumP;
  int pDone;
  unsigned nCand;
};

// Stream one logits row through LDS with double-buffered async global->LDS DMA.
// fn(value, global_index) is called for every element by the thread that owns it.
template <typename F>
__device__ __forceinline__ void stream_row(const float* __restrict__ row,
                                           float (*stage)[CH], int tid, F fn) {
  const int NCH = (VSZ + CH - 1) / CH;
  auto issue = [&](int c) {
    int off = c * CH + tid * 4;
    if (off < VSZ) {
      uint32_t lds = (uint32_t)(uintptr_t)(&stage[c & 1][tid * 4]);
      unsigned long long ga = (unsigned long long)(uintptr_t)(row + off);
      asm volatile("global_load_async_to_lds_b128 %0, %1, off"
                   :: "v"(lds), "v"(ga) : "memory");
    }
  };
  issue(0);
  for (int c = 0; c < NCH; ++c) {
    if (c + 1 < NCH) {
      issue(c + 1);
      asm volatile("s_wait_asynccnt 1" ::: "memory");   // chunk c landed
    } else {
      asm volatile("s_wait_asynccnt 0" ::: "memory");
    }
    __syncthreads();
    int base = c * CH + tid * 4;
    if (base < VSZ) {
      const float* b = &stage[c & 1][tid * 4];
#pragma unroll
      for (int k = 0; k < 4; ++k) fn(b[k], base + k);
    }
    __syncthreads();   // buffer c&1 free for reuse at next iteration's issue
  }
}

__global__ __launch_bounds__(BD)
void sampler_kernel(const float* __restrict__ logits,
                    const float* __restrict__ temps,
                    const int*   __restrict__ topks,
                    const float* __restrict__ topps,
                    const float* __restrict__ minps,
                    const float* __restrict__ us,
                    const int*   __restrict__ needminp,
                    void* __restrict__ out) {
  __shared__ SMem sm;
  const int row = blockIdx.x;
  const int tid = threadIdx.x;
  const float* rowp = logits + (size_t)row * VSZ;

  const float T    = temps[row];
  const float lsc  = (1.0f / T) * 1.4426950408889634f;   // log2(e)/T
  const float topp = topps[row];

  // ---------------- Pass A: online logsumexp (max + denominator) ----------
  float m = -3.402823466e38f, s = 0.0f;
  stream_row(rowp, sm.stage, tid, [&](float v, int gi) {
    (void)gi;
    if (v > m) { s = s * exp2f((m - v) * lsc); m = v; s += 1.0f; }
    else       { s += exp2f((v - m) * lsc); }
  });
  sm.redM[tid] = m; sm.redS[tid] = s;
  __syncthreads();
  for (int st = BD / 2; st > 0; st >>= 1) {
    if (tid < st) {
      float m1 = sm.redM[tid],      s1 = sm.redS[tid];
      float m2 = sm.redM[tid + st], s2 = sm.redS[tid + st];
      float M = fmaxf(m1, m2);
      sm.redM[tid] = M;
      sm.redS[tid] = s1 * exp2f((m1 - M) * lsc) + s2 * exp2f((m2 - M) * lsc);
    }
    __syncthreads();
  }
  if (tid == 0) {
    sm.Lmax = sm.redM[0]; sm.Dsum = sm.redS[0]; sm.invD = 1.0f / sm.redS[0];
    sm.prefK = 0u; sm.prefP = 0u; sm.rK = 64u;
    sm.Aout = 0.0f; sm.pDone = 0; sm.sumP = 1.0f;
  }
  __syncthreads();
  const float Lmax = sm.Lmax, invD = sm.invD;

  // ---- 4-level radix descent: top-64 key (counts) + top-p boundary (mass) ----
  for (int lvl = 0; lvl < 4; ++lvl) {
    if (tid < 256) { sm.cntK[tid] = 0u; sm.cntP[tid] = 0u; sm.wP[tid] = 0.0f; }
    __syncthreads();
    const unsigned pk = sm.prefK, pp = sm.prefP;
    const int pdone = sm.pDone;
    const int shift = 24 - 8 * lvl;

    stream_row(rowp, sm.stage, tid, [&](float v, int gi) {
      (void)gi;
      uint32_t key = fkey(v);
      uint32_t hi  = (lvl == 0) ? 0u : (key >> (32 - 8 * lvl));
      uint32_t bin = (key >> shift) & 255u;
      if (lvl == 0 || hi == pk) atomicAdd(&sm.cntK[bin], 1u);
      if (!pdone && (lvl == 0 || hi == pp)) {
        float p = exp2f((v - Lmax) * lsc) * invD;
        atomicAdd(&sm.wP[bin], p);
        atomicAdd(&sm.cntP[bin], 1u);
      }
    });

    if (tid == 0) {
      // --- top-64 count select ---
      unsigned rK = sm.rK, cum = 0u; int bK = 0;
      for (int b = 255; b >= 0; --b) {
        unsigned nc = cum + sm.cntK[b];
        if (nc >= rK) { bK = b; break; }
        cum = nc;
      }
      sm.rK = rK - cum;
      sm.prefK = (sm.prefK << 8) | (unsigned)bK;
      // --- top-p mass select ---
      if (!sm.pDone) {
        float acc = sm.Aout; int bP = -1;
        for (int b = 255; b >= 0; --b) {
          float na = acc + sm.wP[b];
          if (na > topp) { bP = b; break; }
          acc = na;
        }
        if (bP < 0) {                       // everything kept
          sm.pDone = 1; sm.sumP = acc;
        } else {
          sm.prefP = (sm.prefP << 8) | (unsigned)bP;
          sm.Aout = acc;
          if (lvl == 3) {                   // exact boundary value reached
            float vb = exp2f((funkey(sm.prefP) - Lmax) * lsc) * invD;
            unsigned cnt = sm.cntP[bP]; if (cnt == 0u) cnt = 1u;
            unsigned mm;
            if (vb > 0.0f) {
              float q = floorf((topp - acc) / vb);
              mm = (unsigned)q + 1u;
              if (mm < 1u) mm = 1u;
              if (mm > cnt) mm = cnt;
            } else mm = cnt;
            sm.sumP = acc + (float)mm * vb;
            sm.pDone = 1;
          }
        }
      }
    }
    __syncthreads();
  }

  // ---------------- Gather all elements with key >= key64 ------------------
  if (tid == 0) sm.nCand = 0u;
  sm.skey[tid] = 0ull;                       // padding sinks to the end
  __syncthreads();
  const unsigned key64 = sm.prefK;
  stream_row(rowp, sm.stage, tid, [&](float v, int gi) {
    uint32_t key = fkey(v);
    if (key >= key64) {
      unsigned slot = atomicAdd(&sm.nCand, 1u);
      if (slot < (unsigned)BD)
        sm.skey[slot] = ((unsigned long long)key << 32) |
                        (unsigned long long)(~(uint32_t)gi);
    }
  });

  // ---- bitonic sort, descending on (key, ~idx): prob desc, index asc ------
  for (int k = 2; k <= BD; k <<= 1) {
    for (int j = k >> 1; j > 0; j >>= 1) {
      int i = tid, ixj = i ^ j;
      if (ixj > i) {
        unsigned long long a = sm.skey[i], b = sm.skey[ixj];
        bool up = ((i & k) == 0);
        bool sw = up ? (a < b) : (a > b);
        if (sw) { sm.skey[i] = b; sm.skey[ixj] = a; }
      }
      __syncthreads();
    }
  }

  // ---------------- thread 0: replay reference semantics over top-64 -------
  if (tid == 0) {
    const int   topk = topks[row];
    const float minp = minps[row];
    const float uu   = us[row];
    const int   nmp  = needminp[0];
    const float sumP = sm.sumP;

    float ps[64]; int ord[64]; float filt[64];
    float cs = 0.0f;
    for (int j = 0; j < 64; ++j) {
      unsigned long long e = sm.skey[j];
      float l = funkey((uint32_t)(e >> 32));
      float p = exp2f((l - Lmax) * lsc) * invD;
      ps[j]  = p;
      ord[j] = (int)(~(uint32_t)(e & 0xFFFFFFFFull));
      cs += p;
      float f = p;
      if (j >= topk) f = 0.0f;               // ranks < top_k
      if (cs - p > topp) f = 0.0f;           // csum - probs_sort > top_p
      filt[j] = f;
    }
    if (nmp) {
      float th = filt[0] * minp;
      for (int j = 0; j < 64; ++j) if (filt[j] < th) filt[j] = 0.0f;
    }
    float total = 0.0f;
    for (int j = 0; j < 64; ++j) total += filt[j];
    float ut = uu * total;
    float c2 = 0.0f; int idx = 0;
    for (int j = 0; j < 64; ++j) { c2 += filt[j]; if (c2 < ut) idx++; }
    if (idx > 63) idx = 63;
    if (idx < 0)  idx = 0;
    int   token = ord[idx];
    float ptok  = ps[idx];
    float pn    = ptok / sumP;               // top-p renormalized prob
    float lp;
    if (pn > 0.0f) lp = logf(fmaxf(pn, 1.17549435e-38f));
    else           lp = -3.402823466e38f;

    ((int*)out)[row]          = token;       // token_ids (int32)
    ((float*)out)[BSZ + row]  = lp;          // next_token_logprobs (f32)
  }
}

extern "C" void kernel_launch(void* const* d_in, const int* in_sizes, int n_in,
                              void* d_out, int out_size, void* d_ws, size_t ws_size,
                              hipStream_t stream) {
  (void)in_sizes; (void)n_in; (void)out_size; (void)d_ws; (void)ws_size;
  const float* logits   = (const float*)d_in[0];
  const float* temps    = (const float*)d_in[1];
  const int*   topks    = (const int*)  d_in[2];
  const float* toppsv   = (const float*)d_in[3];
  const float* minpsv   = (const float*)d_in[4];
  const float* uv       = (const float*)d_in[5];
  const int*   needminp = (const int*)  d_in[6];
  sampler_kernel<<<BSZ, BD, 0, stream>>>(logits, temps, topks, toppsv, minpsv,
                                         uv, needminp, d_out);
}